// forward_warp_padded_76785425318152
// MI455X (gfx1250) — compile-verified
//
#include <hip/hip_runtime.h>

// Problem constants (match reference setup_inputs / PAD)
#define B_   8
#define C_   3
#define H_   544
#define W_   960
#define P_   20
#define HP_  (H_ + 2 * P_)   // 584
#define WP_  (W_ + 2 * P_)   // 1000
#define HW_  (H_ * W_)       // 522240 per-channel plane

// Tiling: each block owns a TY x TX tile of the padded source plane and an
// LDS accumulation window of (TY+2D+1) x (TX+2D+1) output pixels x 3 channels.
#define TY_   32
#define TX_   64
#define D_    6                     // halo: covers |flow| <= ~6 (6 sigma)
#define RY_   (TY_ + 2 * D_ + 1)    // 45
#define RX_   (TX_ + 2 * D_ + 1)    // 77
#define RYX_  (RY_ * RX_)           // 3465
#define ACCN_ (C_ * RYX_)           // 10395 floats = 41580 B
#define NT_   256
#define RPT_  (TY_ * TX_ / NT_)     // 8 pixels per thread

#if __has_builtin(__builtin_amdgcn_global_load_async_to_lds_b64)
#define HAVE_ASYNC_LDS 1
#else
#define HAVE_ASYNC_LDS 0
#endif

#if HAVE_ASYNC_LDS
typedef int v2i_ __attribute__((vector_size(8)));              // <2 x i32> = 64-bit payload
typedef __attribute__((address_space(1))) v2i_* g1v2_t;        // global (AS1) pointer
typedef __attribute__((address_space(3))) v2i_* l3v2_t;        // LDS (AS3) pointer
#endif

__device__ __forceinline__ void gadd(float* p, float v) {
    // relaxed device-scope fp32 atomic -> global_atomic_add_f32 (no return)
    __hip_atomic_fetch_add(p, v, __ATOMIC_RELAXED, __HIP_MEMORY_SCOPE_AGENT);
}

// ---------------------------------------------------------------------------
// Zero-fill of d_out (accumulator must start at 0 every call).
// ---------------------------------------------------------------------------
__global__ __launch_bounds__(NT_) void zero_f4(float4* __restrict__ out, int n4) {
    int i = blockIdx.x * NT_ + threadIdx.x;
    if (i < n4) out[i] = make_float4(0.f, 0.f, 0.f, 0.f);
}

// ---------------------------------------------------------------------------
// Tiled forward bilinear splat with LDS accumulation.
//   padded source pixel (y,x) -> output dest (floor(y+flow.y), floor(x+flow.x))
// (the +P shift of the padded flow and the -P crop cancel), so destinations
// are local: accumulate in an LDS window, flush once; rare outliers fall back
// to direct global atomics (exact).
// ---------------------------------------------------------------------------
__global__ __launch_bounds__(NT_) void fwarp_tile(const float* __restrict__ im0,
                                                  const float* __restrict__ flow,
                                                  float* __restrict__ out) {
    __shared__ float acc[ACCN_];
    __shared__ float sflow[TY_ * TX_ * 2];

    const int b   = blockIdx.z;
    const int ty0 = blockIdx.y * TY_;
    const int tx0 = blockIdx.x * TX_;
    const int tid = threadIdx.x;
    const int lx  = tid & (TX_ - 1);
    const int lyb = tid >> 6;                  // 0..3, row stride 4

    // ---- zero the LDS accumulator ----
    for (int i = tid; i < ACCN_; i += NT_) acc[i] = 0.f;

    // ---- stage this thread's flow values into LDS (async path on CDNA5) ----
#pragma unroll
    for (int r = 0; r < RPT_; ++r) {
        int ly = lyb + r * (NT_ / TX_);
        int y  = ty0 + ly;
        int x  = tx0 + lx;
        int li = (ly * TX_ + lx) * 2;
        int yf = y - P_, xf = x - P_;
        bool inflow = ((unsigned)yf < (unsigned)H_) && ((unsigned)xf < (unsigned)W_) &&
                      (y < HP_) && (x < WP_);
        if (inflow) {
            const float* gp = flow + (size_t)((b * H_ + yf) * W_ + xf) * 2;
#if HAVE_ASYNC_LDS
            __builtin_amdgcn_global_load_async_to_lds_b64(
                (g1v2_t)(unsigned long long)(uintptr_t)gp,
                (l3v2_t)(unsigned)(uintptr_t)(sflow + li),
                0, 0);
#else
            const float2 f = *(const float2*)gp;
            sflow[li]     = f.x;
            sflow[li + 1] = f.y;
#endif
        } else {
            sflow[li]     = 0.f;   // zero-padded flow ring (the +P is added later)
            sflow[li + 1] = 0.f;
        }
    }
#if HAVE_ASYNC_LDS
#if __has_builtin(__builtin_amdgcn_s_wait_asynccnt)
    __builtin_amdgcn_s_wait_asynccnt(0);
#else
    asm volatile("s_wait_asynccnt 0x0" ::: "memory");
#endif
    asm volatile("" ::: "memory");
#endif
    __syncthreads();   // acc zeros visible to all before splatting

    const int oy0 = ty0 - D_;   // output coords of LDS window origin
    const int ox0 = tx0 - D_;
    float* outb = out + (size_t)b * (C_ * HW_);

    // ---- splat ----
#pragma unroll
    for (int r = 0; r < RPT_; ++r) {
        int ly = lyb + r * (NT_ / TX_);
        int y  = ty0 + ly;
        int x  = tx0 + lx;
        if (y >= HP_ || x >= WP_) continue;

        int li = (ly * TX_ + lx) * 2;
        float flx = sflow[li];
        float fly = sflow[li + 1];

        // padded-space destination + validity (full 2x2 footprint inside frame)
        float dxp = (float)x + flx + (float)P_;
        float dyp = (float)y + fly + (float)P_;
        float x0f = floorf(dxp);
        float y0f = floorf(dyp);
        if (!(x0f >= 0.f && x0f <= (float)(WP_ - 2) &&
              y0f >= 0.f && y0f <= (float)(HP_ - 2)))
            continue;
        float ax = dxp - x0f;
        float ay = dyp - y0f;
        int Xo0 = (int)x0f - P_;    // NW corner in cropped-output coords
        int Yo0 = (int)y0f - P_;

        // source value via reflect indexing (P < H, P < W -> single bounce)
        int yf = y - P_, xf = x - P_;
        int yr = (yf < 0) ? -yf : ((yf >= H_) ? (2 * H_ - 2 - yf) : yf);
        int xr = (xf < 0) ? -xf : ((xf >= W_) ? (2 * W_ - 2 - xf) : xf);
        const float* imb = im0 + (size_t)b * (C_ * HW_) + yr * W_ + xr;
        float v0 = imb[0];
        float v1 = imb[HW_];
        float v2 = imb[2 * HW_];

        float wx0 = 1.f - ax, wy0 = 1.f - ay;
        const float ws[4] = { wx0 * wy0, ax * wy0, wx0 * ay, ax * ay };
#pragma unroll
        for (int k = 0; k < 4; ++k) {
            float w = ws[k];
            if (w == 0.f) continue;                 // exact no-op in reference
            int Yo = Yo0 + (k >> 1);
            int Xo = Xo0 + (k & 1);
            int wr = Yo - oy0;
            int wc = Xo - ox0;
            if ((unsigned)wr < (unsigned)RY_ && (unsigned)wc < (unsigned)RX_) {
                int a = wr * RX_ + wc;
                atomicAdd(&acc[a],            w * v0);   // ds_add_f32
                atomicAdd(&acc[a + RYX_],     w * v1);
                atomicAdd(&acc[a + 2 * RYX_], w * v2);
            } else if ((unsigned)Yo < (unsigned)H_ && (unsigned)Xo < (unsigned)W_) {
                // rare outlier (|flow| > D): exact global fallback
                float* o = outb + Yo * W_ + Xo;
                gadd(o,           w * v0);
                gadd(o + HW_,     w * v1);
                gadd(o + 2 * HW_, w * v2);
            }
        }
    }

    __syncthreads();

    // ---- flush LDS window to global (coalesced along wc) ----
    for (int i = tid; i < ACCN_; i += NT_) {
        int c   = i / RYX_;
        int rem = i - c * RYX_;
        int wr  = rem / RX_;
        int wc  = rem - wr * RX_;
        int Yo  = oy0 + wr;
        int Xo  = ox0 + wc;
        float v = acc[i];
        if (v != 0.f && (unsigned)Yo < (unsigned)H_ && (unsigned)Xo < (unsigned)W_)
            gadd(outb + (size_t)c * HW_ + Yo * W_ + Xo, v);
    }
}

extern "C" void kernel_launch(void* const* d_in, const int* in_sizes, int n_in,
                              void* d_out, int out_size, void* d_ws, size_t ws_size,
                              hipStream_t stream) {
    const float* im0  = (const float*)d_in[0];   // [8,3,544,960] f32
    const float* flow = (const float*)d_in[1];   // [8,544,960,2] f32
    float*       out  = (float*)d_out;           // [8,3,544,960] f32

    // 1) zero the accumulator output (out_size = 12,533,760, divisible by 4)
    int n4 = out_size / 4;
    zero_f4<<<(n4 + NT_ - 1) / NT_, NT_, 0, stream>>>((float4*)out, n4);

    // 2) tiled splat: grid = (ceil(WP/TX), ceil(HP/TY), B) = (16, 19, 8)
    dim3 grid((WP_ + TX_ - 1) / TX_, (HP_ + TY_ - 1) / TY_, B_);
    fwarp_tile<<<grid, NT_, 0, stream>>>(im0, flow, out);
}